// MLPKAN_9131100472043
// MI455X (gfx1250) — compile-verified
//
#include <hip/hip_runtime.h>
#include <hip/hip_bf16.h>

typedef __attribute__((ext_vector_type(2))) float v2f;
typedef __attribute__((ext_vector_type(8))) float v8f;

// One KAN layer: X[B,I] -> Y[B,O].
// Per connection (i,o): y = w3 . relu(W2 . relu(x*w1 + b1) + b2); Y[b,o] = sum_i y + sum_i b3.
// One wave (32 threads) handles a 16-row batch tile. All matrix math via
// V_WMMA_F32_16X16X4_F32 (K=4 == H0 == H1), biases injected through the C operand.
template <int I, int O>
__global__ __launch_bounds__(32) void kan_layer_wmma(
    const float* __restrict__ X,
    const float* __restrict__ W1, const float* __restrict__ B1,
    const float* __restrict__ W2, const float* __restrict__ B2,
    const float* __restrict__ W3, const float* __restrict__ B3,
    float* __restrict__ Y) {
  __shared__ float lds[16 * 17];

  const int lane = threadIdx.x & 31;
  const int lo   = lane & 15;   // row (A/B major) or column (C/D) index
  const int hl   = lane >> 4;   // which K/M half this lane's VGPRs carry
  const int b0   = blockIdx.x * 16;
  const int oSubLane = lo >> 2; // column -> o_sub within group of 4
  const int hLane    = lo & 3;  // column -> h (or k) within o_sub

  constexpr int NBLK = O / 16;
  v8f acc[NBLK];
  for (int bk = 0; bk < NBLK; ++bk)
#pragma unroll
    for (int r = 0; r < 8; ++r) acc[bk][r] = 0.0f;

  for (int i = 0; i < I; ++i) {
    // x value for batch row (b0+lo); only the K=0 slot (hl==0, vgpr0) is live.
    const float xv = X[(b0 + lo) * I + i];
    v2f a1;
    a1[0] = hl ? 0.0f : xv;
    a1[1] = 0.0f;

    for (int g = 0; g < O / 4; ++g) {
      // ---- Stage A: h1 = relu(x*w1 + b1), 16 cols = (o_sub,k) --------------
      const int pbase = (i * O + 4 * g) * 4 + lo;  // [I,O,4] flat: i*O*4 + 16g + lo
      const float w1v = W1[pbase];
      const float b1v = B1[pbase];
      v2f bop;
      bop[0] = hl ? 0.0f : w1v;  // only B row K=0 nonzero
      bop[1] = 0.0f;
      v8f c1;
#pragma unroll
      for (int r = 0; r < 8; ++r) c1[r] = b1v;  // bias broadcast down rows
      v8f h1 = __builtin_amdgcn_wmma_f32_16x16x4_f32(false, a1, false, bop,
                                                     (short)0, c1, false, false);
#pragma unroll
      for (int r = 0; r < 8; ++r) h1[r] = fmaxf(h1[r], 0.0f);

      // ---- transpose 1: D-layout -> A-layout through LDS -------------------
      asm volatile("" ::: "memory");
#pragma unroll
      for (int r = 0; r < 8; ++r) lds[(r + 8 * hl) * 17 + lo] = h1[r];
      asm volatile("s_wait_dscnt 0" ::: "memory");

      // ---- Stage B: h2 = relu(block-diag W2 * h1 + b2) ---------------------
      const float b2v = B2[pbase];
      v8f h2;
#pragma unroll
      for (int r = 0; r < 8; ++r) h2[r] = b2v;
#pragma unroll
      for (int os = 0; os < 4; ++os) {
        v2f a2;
        a2[0] = lds[lo * 17 + 4 * os + 2 * hl];
        a2[1] = lds[lo * 17 + 4 * os + 2 * hl + 1];
        // B2[k][c'] = w2[i, 4g+os, h(c'), k] iff o_sub(c') == os
        const int w2idx = (((i * O + 4 * g + os) * 4) + hLane) * 4 + 2 * hl;
        const float sel = (oSubLane == os) ? 1.0f : 0.0f;
        v2f b2op;
        b2op[0] = sel * W2[w2idx];
        b2op[1] = sel * W2[w2idx + 1];
        h2 = __builtin_amdgcn_wmma_f32_16x16x4_f32(false, a2, false, b2op,
                                                   (short)0, h2, false, false);
      }
#pragma unroll
      for (int r = 0; r < 8; ++r) h2[r] = fmaxf(h2[r], 0.0f);

      // ---- transpose 2 -----------------------------------------------------
      asm volatile("" ::: "memory");
#pragma unroll
      for (int r = 0; r < 8; ++r) lds[(r + 8 * hl) * 17 + lo] = h2[r];
      asm volatile("s_wait_dscnt 0" ::: "memory");

      // ---- Stage C: acc[:, o mod 16] += sum_h h2[:,(o,h)] * w3[i,o,h] ------
      const int blk = g >> 2;
#pragma unroll
      for (int os = 0; os < 4; ++os) {
        v2f a3;
        a3[0] = lds[lo * 17 + 4 * os + 2 * hl];
        a3[1] = lds[lo * 17 + 4 * os + 2 * hl + 1];
        const int o = 4 * g + os;
        const float sel = (lo == (o & 15)) ? 1.0f : 0.0f;  // single target column
        const int w3idx = (i * O + o) * 4 + 2 * hl;
        v2f b3op;
        b3op[0] = sel * W3[w3idx];
        b3op[1] = sel * W3[w3idx + 1];
        acc[blk] = __builtin_amdgcn_wmma_f32_16x16x4_f32(
            false, a3, false, b3op, (short)0, acc[blk], false, false);
      }
    }
  }

  // ---- epilogue: add sum_i b3[i,o], store -----------------------------------
  for (int bk = 0; bk < NBLK; ++bk) {
    const int o = 16 * bk + lo;
    float b3s = 0.0f;
    for (int i = 0; i < I; ++i) b3s += B3[i * O + o];
#pragma unroll
    for (int r = 0; r < 8; ++r)
      Y[(b0 + r + 8 * hl) * O + o] = acc[bk][r] + b3s;
  }
}

extern "C" void kernel_launch(void* const* d_in, const int* in_sizes, int n_in,
                              void* d_out, int out_size, void* d_ws, size_t ws_size,
                              hipStream_t stream) {
  (void)in_sizes; (void)n_in; (void)out_size; (void)ws_size;
  constexpr int B = 8192;

  const float* x    = (const float*)d_in[0];
  const float* w1_0 = (const float*)d_in[1];
  const float* b1_0 = (const float*)d_in[2];
  const float* w2_0 = (const float*)d_in[3];
  const float* b2_0 = (const float*)d_in[4];
  const float* w3_0 = (const float*)d_in[5];
  const float* b3_0 = (const float*)d_in[6];
  const float* w1_1 = (const float*)d_in[7];
  const float* b1_1 = (const float*)d_in[8];
  const float* w2_1 = (const float*)d_in[9];
  const float* b2_1 = (const float*)d_in[10];
  const float* w3_1 = (const float*)d_in[11];
  const float* b3_1 = (const float*)d_in[12];

  float* h   = (float*)d_ws;       // [B,32] layer-0 output (1 MB scratch)
  float* out = (float*)d_out;      // [B,16]

  dim3 blk(32);
  dim3 grd(B / 16);
  kan_layer_wmma<32, 32><<<grd, blk, 0, stream>>>(x, w1_0, b1_0, w2_0, b2_0,
                                                  w3_0, b3_0, h);
  kan_layer_wmma<32, 16><<<grd, blk, 0, stream>>>(h, w1_1, b1_1, w2_1, b2_1,
                                                  w3_1, b3_1, out);
}